// LRU_13477607375571
// MI455X (gfx1250) — compile-verified
//
#include <hip/hip_runtime.h>
#include <math.h>

// ---------------------------------------------------------------------------
// LRU (bidirectional complex linear recurrence) for MI455X / gfx1250.
//   T=16384 (L), N=512 states, H=512.
//
//   1) prep:   Lambda = exp(-exp(nu)+i*exp(theta)); store CONJUGATE Lambda
//              (see below) and gamma = sqrt(1-|Lambda|^2)
//   2) GEMM1:  (Bu_re, -Bu_im) = gamma[n] * (x @ B^T), written to fwd & bwd
//              planes.  v_wmma_f32_16x16x4_f32; async global->LDS double
//              buffered staging (ASYNCcnt).
//   3) scan:   chunked (CHUNK=256) scans of the conjugate system: planes end
//              up holding (s_re, -s_im), so the output GEMM is all-plus:
//              y = fwd_re*C_re + (-fwd_im)*C_im + bwd_re*C_re2 + (-bwd_im)*C_im2
//   4) GEMM2:  y = planes @ [C planes] + D[h]*x   (K=2048, WMMA f32, async)
//
// Roofline: ~52 GFLOP fp32 vs ~250 MB HBM traffic -> fp32-WMMA compute bound;
// v_wmma_f32_16x16x4_f32 is the precision-correct matrix path on CDNA5.
// ---------------------------------------------------------------------------

typedef float v2f __attribute__((ext_vector_type(2)));
typedef float v8f __attribute__((ext_vector_type(8)));

#define T_LEN   16384
#define N_ST    512
#define H_DIM   512
#define CHUNK   256
#define NCHUNK  (T_LEN / CHUNK)          // 64
#define LN      ((size_t)T_LEN * N_ST)   // elems per plane

#define WMMA_F32(A, B, C) \
  __builtin_amdgcn_wmma_f32_16x16x4_f32(false, (A), false, (B), (short)0, (C), false, false)

// Async copy of one 16B chunk, global -> LDS (contiguous both sides).
// Generic pointers to __shared__ carry the LDS byte offset in their low 32
// bits on gfx1250, which is exactly what the VDST operand wants.
__device__ __forceinline__ void async_cp16(const float* __restrict__ g, void* l) {
  unsigned lo = (unsigned)(size_t)l;
  asm volatile("global_load_async_to_lds_b128 %0, %1, off"
               :: "v"(lo), "v"(g) : "memory");
}
__device__ __forceinline__ void async_wait_all() {
  asm volatile("s_wait_asynccnt 0x0" ::: "memory");
}

// ---------------------------------------------------------------- prep
__global__ __launch_bounds__(512)
void k_prep(const float* __restrict__ theta_log, const float* __restrict__ nu_log,
            float* __restrict__ lam_re, float* __restrict__ lam_im,
            float* __restrict__ gam) {
  int n = threadIdx.x;
  if (n < N_ST) {
    float nu  = __expf(nu_log[n]);
    float th  = __expf(theta_log[n]);
    float mag = __expf(-nu);
    lam_re[n] = mag * __cosf(th);
    lam_im[n] = -mag * __sinf(th);     // CONJUGATE: scans run on conj(Lambda)
    gam[n]    = sqrtf(fmaxf(0.f, 1.f - mag * mag));
  }
}

// ---------------------------------------------------------------- GEMM1
// Block: 128 threads (4 waves), tile 128(M) x 64(N), BK=16, double buffered.
// Wave computes 32(M) x 64(N): 2 m-tiles x 4 n-tiles x {re,im}.
__global__ __launch_bounds__(128)
void k_gemm1(const float* __restrict__ x, const float* __restrict__ Bre,
             const float* __restrict__ Bim, const float* __restrict__ gam,
             float* __restrict__ planes) {
  // k-contiguous layouts, even pitch (18) -> aligned v2f fragment loads.
  __shared__ float As [2][128][18];
  __shared__ float BsR[2][64][18];
  __shared__ float BsI[2][64][18];

  const int tid  = threadIdx.x;
  const int wave = tid >> 5;
  const int lane = tid & 31;
  const int m0   = blockIdx.x * 128;
  const int n0   = blockIdx.y * 64;

  v8f accR[2][4] = {};
  v8f accI[2][4] = {};

  const int mrow0 = (wave << 5) + (lane & 15);  // m-tile 0 row for this lane
  const int koff  = (lane >> 4) << 1;           // lanes 16-31 hold K+2,K+3
  const int bc    = lane & 15;                  // B/N column within a 16-tile

  // issue all async chunk copies for one k-tile into buffer `b`
  auto issue = [&](int b, int k0) {
    #pragma unroll
    for (int c = tid; c < 512; c += 128) {      // A: 128x16
      int row = c >> 2, c4 = (c & 3) << 2;
      async_cp16(x + (size_t)(m0 + row) * H_DIM + k0 + c4, &As[b][row][c4]);
    }
    #pragma unroll
    for (int c = tid; c < 256; c += 128) {      // B_re / B_im: 64x16
      int nn = c >> 2, c4 = (c & 3) << 2;
      async_cp16(Bre + (size_t)(n0 + nn) * H_DIM + k0 + c4, &BsR[b][nn][c4]);
      async_cp16(Bim + (size_t)(n0 + nn) * H_DIM + k0 + c4, &BsI[b][nn][c4]);
    }
  };

  issue(0, 0);
  async_wait_all();
  __syncthreads();

  int buf = 0;
  const int NT = H_DIM / 16;
  for (int kt = 0; kt < NT; ++kt) {
    if (kt + 1 < NT) issue(buf ^ 1, (kt + 1) * 16);
    #pragma unroll
    for (int kk = 0; kk < 16; kk += 4) {
      v2f a0 = *(const v2f*)&As[buf][mrow0][kk + koff];
      v2f a1 = *(const v2f*)&As[buf][mrow0 + 16][kk + koff];
      #pragma unroll
      for (int nt = 0; nt < 4; ++nt) {
        v2f br = *(const v2f*)&BsR[buf][(nt << 4) + bc][kk + koff];
        v2f bi = *(const v2f*)&BsI[buf][(nt << 4) + bc][kk + koff];
        accR[0][nt] = WMMA_F32(a0, br, accR[0][nt]);
        accI[0][nt] = WMMA_F32(a0, bi, accI[0][nt]);
        accR[1][nt] = WMMA_F32(a1, br, accR[1][nt]);
        accI[1][nt] = WMMA_F32(a1, bi, accI[1][nt]);
      }
    }
    async_wait_all();
    __syncthreads();
    buf ^= 1;
  }

  // epilogue: scale by gamma[n]; im planes get NEGATED Bu_im (conjugate scan);
  // replicate into fwd & bwd planes for in-place scans.
  float* fwdR = planes;            float* fwdI = planes + LN;
  float* bwdR = planes + 2 * LN;   float* bwdI = planes + 3 * LN;
  const int rowoff = (lane >> 4) << 3;   // 0 or 8
  #pragma unroll
  for (int mt = 0; mt < 2; ++mt) {
    #pragma unroll
    for (int nt = 0; nt < 4; ++nt) {
      const int n = n0 + (nt << 4) + bc;
      const float g = gam[n];
      #pragma unroll
      for (int r = 0; r < 8; ++r) {
        const int m = m0 + (wave << 5) + (mt << 4) + rowoff + r;
        const size_t o = (size_t)m * N_ST + n;
        const float vr =  g * accR[mt][nt][r];
        const float vi = -g * accI[mt][nt][r];
        fwdR[o] = vr; fwdI[o] = vi; bwdR[o] = vr; bwdI[o] = vi;
      }
    }
  }
}

// ---------------------------------------------------------------- scan pass 1
__global__ __launch_bounds__(256)
void k_scan_local(float* __restrict__ planes, float* __restrict__ end_re,
                  float* __restrict__ end_im, const float* __restrict__ lam_re,
                  const float* __restrict__ lam_im) {
  const int gid   = blockIdx.x * 256 + threadIdx.x;
  const int n     = gid & (N_ST - 1);
  const int rest  = gid >> 9;
  const int chunk = rest & (NCHUNK - 1);
  const int dir   = rest >> 6;

  float* pre = planes + (dir ? 2 * LN : 0);
  float* pim = pre + LN;
  const float lr = lam_re[n], li = lam_im[n];
  float sr = 0.f, si = 0.f;
  int l = dir ? (T_LEN - 1 - chunk * CHUNK) : chunk * CHUNK;
  const int step = dir ? -1 : 1;
  for (int t = 0; t < CHUNK; ++t, l += step) {
    const size_t o = (size_t)l * N_ST + n;
    const float br = pre[o], bi = pim[o];
    const float nsr = fmaf(lr, sr, fmaf(-li, si, br));
    const float nsi = fmaf(lr, si, fmaf( li, sr, bi));
    sr = nsr; si = nsi;
    pre[o] = sr; pim[o] = si;
  }
  const int eo = (dir * NCHUNK + chunk) * N_ST + n;
  end_re[eo] = sr; end_im[eo] = si;
}

// ---------------------------------------------------------------- scan pass 2
// conj(Lambda)^CHUNK = mag^CHUNK * (cos(CHUNK*th), -sin(CHUNK*th))
__global__ __launch_bounds__(256)
void k_scan_carry(const float* __restrict__ end_re, const float* __restrict__ end_im,
                  float* __restrict__ car_re, float* __restrict__ car_im,
                  const float* __restrict__ theta_log, const float* __restrict__ nu_log) {
  const int gid = blockIdx.x * 256 + threadIdx.x;
  if (gid >= 2 * N_ST) return;
  const int n = gid & (N_ST - 1);
  const int dir = gid >> 9;
  const float nu  = __expf(nu_log[n]);
  const float th  = __expf(theta_log[n]);
  const float mag = __expf(-(float)CHUNK * nu);
  const float ar  =  mag * __cosf((float)CHUNK * th);
  const float ai  = -mag * __sinf((float)CHUNK * th);   // conjugate
  const int base = dir * NCHUNK * N_ST;
  float cr = 0.f, ci = 0.f;
  car_re[base + n] = 0.f; car_im[base + n] = 0.f;
  for (int c = 1; c < NCHUNK; ++c) {
    const float er = end_re[base + (c - 1) * N_ST + n];
    const float ei = end_im[base + (c - 1) * N_ST + n];
    const float pr = fmaf(ar, cr, fmaf(-ai, ci, er));
    const float pi = fmaf(ar, ci, fmaf( ai, cr, ei));
    cr = pr; ci = pi;
    car_re[base + c * N_ST + n] = cr;
    car_im[base + c * N_ST + n] = ci;
  }
}

// ---------------------------------------------------------------- scan pass 3
__global__ __launch_bounds__(256)
void k_scan_apply(float* __restrict__ planes, const float* __restrict__ car_re,
                  const float* __restrict__ car_im, const float* __restrict__ lam_re,
                  const float* __restrict__ lam_im) {
  const int gid   = blockIdx.x * 256 + threadIdx.x;
  const int n     = gid & (N_ST - 1);
  const int rest  = gid >> 9;
  const int chunk = rest & (NCHUNK - 1);
  const int dir   = rest >> 6;
  if (chunk == 0) return;

  float* pre = planes + (dir ? 2 * LN : 0);
  float* pim = pre + LN;
  const float lr = lam_re[n], li = lam_im[n];
  const int co = (dir * NCHUNK + chunk) * N_ST + n;
  const float cr = car_re[co], ci = car_im[co];
  float pr = lr, pi = li;
  int l = dir ? (T_LEN - 1 - chunk * CHUNK) : chunk * CHUNK;
  const int step = dir ? -1 : 1;
  for (int t = 0; t < CHUNK; ++t, l += step) {
    const size_t o = (size_t)l * N_ST + n;
    pre[o] = fmaf(pr, cr, fmaf(-pi, ci, pre[o]));
    pim[o] = fmaf(pr, ci, fmaf( pi, cr, pim[o]));
    const float npr = pr * lr - pi * li;
    const float npi = pr * li + pi * lr;
    pr = npr; pi = npi;
  }
}

// ---------------------------------------------------------------- GEMM2
// y[l,h] = sum_{plane,n} planes[plane][l,n] * C_plane[h,n] + D[h]*x[l,h]
// All-plus because im planes hold -s_im.  K = 2048, double buffered async.
__global__ __launch_bounds__(128)
void k_gemm2(const float* __restrict__ planes,
             const float* __restrict__ Cre, const float* __restrict__ Cim,
             const float* __restrict__ Cre2, const float* __restrict__ Cim2,
             const float* __restrict__ x, const float* __restrict__ D,
             float* __restrict__ y) {
  __shared__ float As[2][128][18];
  __shared__ float Bs[2][64][18];

  const int tid  = threadIdx.x;
  const int wave = tid >> 5;
  const int lane = tid & 31;
  const int m0   = blockIdx.x * 128;
  const int h0   = blockIdx.y * 64;

  v8f acc[2][4] = {};
  const int mrow0 = (wave << 5) + (lane & 15);
  const int koff  = (lane >> 4) << 1;
  const int bc    = lane & 15;

  const float* Csrc[4] = {Cre, Cim, Cre2, Cim2};

  auto issue = [&](int b, int k0) {
    const int plane = k0 >> 9;
    const int kl    = k0 & (N_ST - 1);
    const float* Ap = planes + (size_t)plane * LN;
    const float* Bp = Csrc[plane];
    #pragma unroll
    for (int c = tid; c < 512; c += 128) {      // A: 128x16
      int row = c >> 2, c4 = (c & 3) << 2;
      async_cp16(Ap + (size_t)(m0 + row) * N_ST + kl + c4, &As[b][row][c4]);
    }
    #pragma unroll
    for (int c = tid; c < 256; c += 128) {      // B: 64(h) x 16(k=n)
      int hh = c >> 2, c4 = (c & 3) << 2;
      async_cp16(Bp + (size_t)(h0 + hh) * N_ST + kl + c4, &Bs[b][hh][c4]);
    }
  };

  issue(0, 0);
  async_wait_all();
  __syncthreads();

  int buf = 0;
  const int NT = (4 * N_ST) / 16;   // 128
  for (int kt = 0; kt < NT; ++kt) {
    if (kt + 1 < NT) issue(buf ^ 1, (kt + 1) * 16);
    #pragma unroll
    for (int kk = 0; kk < 16; kk += 4) {
      v2f a0 = *(const v2f*)&As[buf][mrow0][kk + koff];
      v2f a1 = *(const v2f*)&As[buf][mrow0 + 16][kk + koff];
      #pragma unroll
      for (int nt = 0; nt < 4; ++nt) {
        v2f b = *(const v2f*)&Bs[buf][(nt << 4) + bc][kk + koff];
        acc[0][nt] = WMMA_F32(a0, b, acc[0][nt]);
        acc[1][nt] = WMMA_F32(a1, b, acc[1][nt]);
      }
    }
    async_wait_all();
    __syncthreads();
    buf ^= 1;
  }

  const int rowoff = (lane >> 4) << 3;
  #pragma unroll
  for (int mt = 0; mt < 2; ++mt) {
    #pragma unroll
    for (int nt = 0; nt < 4; ++nt) {
      const int h = h0 + (nt << 4) + bc;
      const float dh = D[h];
      #pragma unroll
      for (int r = 0; r < 8; ++r) {
        const int m = m0 + (wave << 5) + (mt << 4) + rowoff + r;
        const size_t o = (size_t)m * H_DIM + h;
        y[o] = fmaf(dh, x[o], acc[mt][nt][r]);
      }
    }
  }
}

// ---------------------------------------------------------------- launcher
extern "C" void kernel_launch(void* const* d_in, const int* in_sizes, int n_in,
                              void* d_out, int out_size, void* d_ws, size_t ws_size,
                              hipStream_t stream) {
  const float* x         = (const float*)d_in[0];
  const float* theta_log = (const float*)d_in[1];
  const float* nu_log    = (const float*)d_in[2];
  const float* B_re      = (const float*)d_in[3];
  const float* B_im      = (const float*)d_in[4];
  const float* C_re      = (const float*)d_in[5];
  const float* C_im      = (const float*)d_in[6];
  const float* C_re2     = (const float*)d_in[7];
  const float* C_im2     = (const float*)d_in[8];
  const float* Dv        = (const float*)d_in[9];
  float* y = (float*)d_out;

  float* w       = (float*)d_ws;
  float* lam_re  = w;
  float* lam_im  = w + 512;
  float* gam     = w + 1024;
  float* planes  = w + 2048;                 // 4 planes of L*N
  float* end_re  = planes + 4 * LN;
  float* end_im  = end_re + 2 * NCHUNK * N_ST;
  float* car_re  = end_im + 2 * NCHUNK * N_ST;
  float* car_im  = car_re + 2 * NCHUNK * N_ST;

  k_prep<<<1, 512, 0, stream>>>(theta_log, nu_log, lam_re, lam_im, gam);

  k_gemm1<<<dim3(T_LEN / 128, N_ST / 64), 128, 0, stream>>>(x, B_re, B_im, gam, planes);

  k_scan_local<<<(2 * NCHUNK * N_ST) / 256, 256, 0, stream>>>(planes, end_re, end_im,
                                                              lam_re, lam_im);
  k_scan_carry<<<(2 * N_ST + 255) / 256, 256, 0, stream>>>(end_re, end_im, car_re, car_im,
                                                           theta_log, nu_log);
  k_scan_apply<<<(2 * NCHUNK * N_ST) / 256, 256, 0, stream>>>(planes, car_re, car_im,
                                                              lam_re, lam_im);

  k_gemm2<<<dim3(T_LEN / 128, H_DIM / 64), 128, 0, stream>>>(planes, C_re, C_im, C_re2,
                                                             C_im2, x, Dv, y);
}